// SAMMultiLevelAssignLoss_28707561407033
// MI455X (gfx1250) — compile-verified
//
#include <hip/hip_runtime.h>

// ---------------------------------------------------------------------------
// SAM multi-level assign loss, MI455X (gfx1250, wave32)
// Pass 1: stream 160MB once (b128 loads), 21 fp32 partials per (b,chunk),
//         wave32 shfl_xor butterflies + V_WMMA_F32_16X16X4_F32 (ones-matrix,
//         layout-independent) cross-wave summation epilogue.
// Pass 2: tiny deterministic finalize (128 threads) -> 3 scalars.
// ---------------------------------------------------------------------------

typedef __attribute__((ext_vector_type(2))) float v2f;
typedef __attribute__((ext_vector_type(8))) float v8f;

#define HW_TOT 65536   // 256*256
#define KLEV   4
#define NB     128
#define CHUNKS 16
#define CHUNK  4096    // HW_TOT / CHUNKS
#define NVAL   21      // [0]=t_sum, then per k: focal,p_sum,inter,m_sum,i2

__device__ __forceinline__ void accum_elem(float x, float t,
    float& f, float& ps, float& inter, float& ms, float& i2) {
  float ax  = __builtin_fabsf(x);
  float e   = __expf(-ax);                       // exp(-|x|)
  float bce = fmaxf(x, 0.0f) - x * t + __logf(1.0f + e);
  float u   = 1.0f - __expf(-bce);
  f += 0.8f * (u * u) * bce;                     // ALPHA=0.8, GAMMA=2
  float inv = 1.0f / (1.0f + e);
  float sig = (x >= 0.0f) ? inv : e * inv;       // sigmoid(x)
  float p   = fminf(fmaxf(sig, 1.0e-4f), 1.0f - 1.0e-4f);
  ps    += p;
  inter += p * t;
  float m = (x >= 0.0f) ? 1.0f : 0.0f;           // MASK_THRESH = 0
  ms += m;
  i2 += m * t;
}

__global__ void __launch_bounds__(256)
sam_partial_kernel(const float* __restrict__ pred,
                   const float* __restrict__ tgt,
                   float* __restrict__ ws) {
  const int c   = blockIdx.x;   // chunk 0..15
  const int b   = blockIdx.y;   // batch 0..127
  const int tid = threadIdx.x;  // 0..255

  const float* tp = tgt  + (size_t)b * HW_TOT + (size_t)c * CHUNK;
  const float* pp = pred + (size_t)b * KLEV * HW_TOT + (size_t)c * CHUNK;

  float acc[NVAL];
  #pragma unroll
  for (int j = 0; j < NVAL; ++j) acc[j] = 0.0f;

  #pragma unroll
  for (int it = 0; it < CHUNK / 1024; ++it) {     // 256 thr * float4 = 1024/iter
    const int off = it * 1024 + tid * 4;
    const float4 tv = *reinterpret_cast<const float4*>(tp + off);
    acc[0] += (tv.x + tv.y) + (tv.z + tv.w);
    #pragma unroll
    for (int k = 0; k < KLEV; ++k) {
      const float4 xv =
          *reinterpret_cast<const float4*>(pp + (size_t)k * HW_TOT + off);
      float* a = &acc[1 + 5 * k];
      accum_elem(xv.x, tv.x, a[0], a[1], a[2], a[3], a[4]);
      accum_elem(xv.y, tv.y, a[0], a[1], a[2], a[3], a[4]);
      accum_elem(xv.z, tv.z, a[0], a[1], a[2], a[3], a[4]);
      accum_elem(xv.w, tv.w, a[0], a[1], a[2], a[3], a[4]);
    }
  }

  // --- wave32 butterfly: every lane ends with the wave sum ---
  #pragma unroll
  for (int j = 0; j < NVAL; ++j) {
    float v = acc[j];
    #pragma unroll
    for (int o = 16; o > 0; o >>= 1) v += __shfl_xor(v, o, 32);
    acc[j] = v;
  }

  __shared__ float sred[8][NVAL];
  const int lane = tid & 31;
  const int wv   = tid >> 5;
  if (lane == 0) {
    #pragma unroll
    for (int j = 0; j < NVAL; ++j) sred[wv][j] = acc[j];
  }
  __syncthreads();

  // --- cross-wave sum (8 partials per value) via WMMA f32 16x16x4 ---
  // A = ones(16x4) => D[m][n] = sum_k B[k][n]; result independent of the
  // exact k<->(vgpr,lane-half) slot mapping; only needs N = lane&15.
  if (tid < 32) {                                  // wave 0 only, EXEC all-1s
    float* wsrow   = ws + ((size_t)b * CHUNKS + c) * NVAL;
    const int half = lane >> 4;                    // 0 or 1
    const int col  = lane & 15;                    // N index

    v2f a; a[0] = 1.0f; a[1] = 1.0f;               // ones A-matrix

    // group 0: values 0..15
    v2f blo, bhi;
    blo[0] = sred[half][col];     blo[1] = sred[2 + half][col];
    bhi[0] = sred[4 + half][col]; bhi[1] = sred[6 + half][col];
    v8f d0 = {};
    d0 = __builtin_amdgcn_wmma_f32_16x16x4_f32(false, a, false, blo,
                                               (short)0, d0, false, false);
    d0 = __builtin_amdgcn_wmma_f32_16x16x4_f32(false, a, false, bhi,
                                               (short)0, d0, false, false);

    // group 1: values 16..20 (pad columns 5..15 with zero)
    const int  vhi = 16 + col;
    const bool ok  = (vhi < NVAL);
    const int  vc  = ok ? vhi : 0;                 // keep LDS index in range
    v2f clo, chi;
    float s0 = sred[half][vc],     s1 = sred[2 + half][vc];
    float s2 = sred[4 + half][vc], s3 = sred[6 + half][vc];
    clo[0] = ok ? s0 : 0.0f; clo[1] = ok ? s1 : 0.0f;
    chi[0] = ok ? s2 : 0.0f; chi[1] = ok ? s3 : 0.0f;
    v8f d1 = {};
    d1 = __builtin_amdgcn_wmma_f32_16x16x4_f32(false, a, false, clo,
                                               (short)0, d1, false, false);
    d1 = __builtin_amdgcn_wmma_f32_16x16x4_f32(false, a, false, chi,
                                               (short)0, d1, false, false);

    if (lane < 16) {
      wsrow[lane] = d0[0];                         // D row M=0, N=lane
      if (ok) wsrow[vhi] = d1[0];
    }
  }
}

__global__ void __launch_bounds__(128)
sam_finalize_kernel(const float* __restrict__ ws,
                    const float* __restrict__ ious,
                    float* __restrict__ out) {
  const int tid = threadIdx.x;                     // == b, 0..127
  float v[NVAL];
  #pragma unroll
  for (int j = 0; j < NVAL; ++j) v[j] = 0.0f;
  for (int ci = 0; ci < CHUNKS; ++ci) {
    const float* p = ws + ((size_t)tid * CHUNKS + ci) * NVAL;
    #pragma unroll
    for (int j = 0; j < NVAL; ++j) v[j] += p[j];
  }

  const float tsum  = v[0];
  const float ratio = tsum * (1.0f / (float)HW_TOT);
  const float lo[4] = {0.00f, 0.01f, 0.09f, 0.25f};
  const float hi[4] = {0.04f, 0.16f, 0.49f, 1.00f};

  float cnt = 0.f, fs = 0.f, dsm = 0.f, ism = 0.f;
  #pragma unroll
  for (int k = 0; k < KLEV; ++k) {
    const float valid = (ratio > lo[k] && ratio < hi[k]) ? 1.0f : 0.0f;
    const float* a = &v[1 + 5 * k];
    const float focal_bk = a[0] * (1.0f / (float)HW_TOT);
    const float dice_bk  = 1.0f - (2.0f * a[2] + 1.0e-4f) / (a[1] + tsum + 1.0e-4f);
    const float iou_gt   = (a[4] + 1.0e-4f) / (a[3] + tsum - a[4] + 1.0e-4f);
    const float dlt      = ious[tid * KLEV + k] - iou_gt;
    cnt += valid;
    fs  += focal_bk * valid;
    dsm += dice_bk * valid;
    ism += dlt * dlt * valid;
  }

  float pf = 0.f, pd = 0.f, pi = 0.f, vb = 0.f;
  if (cnt > 0.0f) {
    const float invc = 1.0f / fmaxf(cnt, 1.0f);
    pf = fs * invc; pd = dsm * invc; pi = ism * invc; vb = 1.0f;
  }

  #pragma unroll
  for (int o = 16; o > 0; o >>= 1) {
    pf += __shfl_xor(pf, o, 32);
    pd += __shfl_xor(pd, o, 32);
    pi += __shfl_xor(pi, o, 32);
    vb += __shfl_xor(vb, o, 32);
  }

  __shared__ float red[4][4];
  const int lane = tid & 31, wv = tid >> 5;
  if (lane == 0) { red[wv][0] = pf; red[wv][1] = pd; red[wv][2] = pi; red[wv][3] = vb; }
  __syncthreads();

  if (tid == 0) {
    float F = 0.f, D = 0.f, I = 0.f, V = 0.f;
    #pragma unroll
    for (int w = 0; w < 4; ++w) { F += red[w][0]; D += red[w][1]; I += red[w][2]; V += red[w][3]; }
    const float invv = 1.0f / fmaxf(V, 1.0f);
    out[0] = 20.0f * ((V > 0.f) ? F * invv : F);   // FOCAL_W
    out[1] = (V > 0.f) ? D * invv : D;             // DICE_W
    out[2] = (V > 0.f) ? I * invv : I;             // IOU_W
  }
}

extern "C" void kernel_launch(void* const* d_in, const int* in_sizes, int n_in,
                              void* d_out, int out_size, void* d_ws, size_t ws_size,
                              hipStream_t stream) {
  (void)in_sizes; (void)n_in; (void)out_size; (void)ws_size;
  const float* pred = (const float*)d_in[0];   // (128,4,256,256)
  const float* ious = (const float*)d_in[1];   // (128,4)
  const float* tgt  = (const float*)d_in[2];   // (128,1,256,256)
  float* ws  = (float*)d_ws;                   // 128*16*21 floats = 172 KB
  float* out = (float*)d_out;                  // 3 scalars

  dim3 grid(CHUNKS, NB);
  sam_partial_kernel<<<grid, 256, 0, stream>>>(pred, tgt, ws);
  sam_finalize_kernel<<<1, 128, 0, stream>>>(ws, ious, out);
}